// RadialTokenizer_87531433492823
// MI455X (gfx1250) — compile-verified
//
#include <hip/hip_runtime.h>
#include <hip/hip_bf16.h>
#include <cmath>

#define IMG_SZ    256
#define HW        (IMG_SZ * IMG_SZ)   // 65536
#define NRINGS    16
#define NB        32
#define NC        3
#define BC        (NB * NC)           // 96
#define TOK       9                   // mean(3) | std(3) | median(3)
#define MAXRING   8192                // largest annulus ~6233 px

typedef float v2f __attribute__((ext_vector_type(2)));
typedef float v8f __attribute__((ext_vector_type(8)));

// ---------------------------------------------------------------------------
// Pass 0: zero the s1/s2 accumulators in workspace (graph-replay safe).
// ---------------------------------------------------------------------------
__global__ void rt_zero_kernel(float* __restrict__ p, int n) {
    int i = blockIdx.x * blockDim.x + threadIdx.x;
    if (i < n) p[i] = 0.0f;
}

// ---------------------------------------------------------------------------
// Pass 1: s1/s2 via V_WMMA_F32_16X16X4_F32.
//   D(16x16) = A(16x4) * B(4x16) + C, accumulated over K = HW in steps of 4.
//   A rows = 16 consecutive (b,c) rows of the image, B cols = 16 rings.
//   A layout: lanes 0-15 -> M=lane, VGPR0/1 = K0/K1; lanes 16-31 -> K2/K3.
//   B layout: lanes 0-15 -> N=lane, VGPR0/1 = K0/K1; lanes 16-31 -> K2/K3.
//   C/D layout: lane<16 -> (M=v, N=lane); lane>=16 -> (M=v+8, N=lane-16).
// One wave per block; grid = (HW/CHUNK) x (BC/16). Partials atomically added.
// ---------------------------------------------------------------------------
__global__ __launch_bounds__(32)
void rt_wmma_moments_kernel(const float* __restrict__ img,
                            const float* __restrict__ mask,
                            float* __restrict__ s1,
                            float* __restrict__ s2,
                            int chunk) {
    const int lane = threadIdx.x;        // 0..31
    const int half = lane >> 4;          // 0 or 1 -> K pair select
    const int mrow = lane & 15;          // A row / B col within tile
    const int tile = blockIdx.y;         // which 16-row bc tile (0..5)

    const size_t arow = (size_t)(tile * 16 + mrow) * HW;  // image (b*3+c) row
    const size_t brow = (size_t)mrow * HW;                // ring row

    const int k0   = blockIdx.x * chunk;
    const int kend = k0 + chunk;

    v8f acc1 = {};
    v8f acc2 = {};

    for (int k = k0; k < kend; k += 4) {
        v2f a  = *(const v2f*)(img  + arow + k + 2 * half);
        v2f bm = *(const v2f*)(mask + brow + k + 2 * half);
        v2f a2 = a * a;
        acc1 = __builtin_amdgcn_wmma_f32_16x16x4_f32(
                   false, a,  false, bm, (short)0, acc1, false, false);
        acc2 = __builtin_amdgcn_wmma_f32_16x16x4_f32(
                   false, a2, false, bm, (short)0, acc2, false, false);
    }

    const int n     = mrow;                 // output column (ring)
    const int mbase = tile * 16 + half * 8; // output row base (b*3+c)
#pragma unroll
    for (int v = 0; v < 8; ++v) {
        atomicAdd(&s1[(mbase + v) * NRINGS + n], acc1[v]);
        atomicAdd(&s2[(mbase + v) * NRINGS + n], acc2[v]);
    }
}

// ---------------------------------------------------------------------------
// Pass 2: per-ring pixel counts (16 blocks, LDS tree reduction).
// ---------------------------------------------------------------------------
__global__ __launch_bounds__(256)
void rt_counts_kernel(const float* __restrict__ mask, float* __restrict__ counts) {
    __shared__ float red[256];
    const int r = blockIdx.x;
    float s = 0.0f;
    for (int i = threadIdx.x; i < HW; i += 256)
        s += mask[(size_t)r * HW + i];
    red[threadIdx.x] = s;
    __syncthreads();
    for (int off = 128; off > 0; off >>= 1) {
        if (threadIdx.x < off) red[threadIdx.x] += red[threadIdx.x + off];
        __syncthreads();
    }
    if (threadIdx.x == 0) counts[r] = red[0];
}

// ---------------------------------------------------------------------------
// Pass 3: mean / std from s1, s2, counts.
// ---------------------------------------------------------------------------
__global__ void rt_finalize_kernel(const float* __restrict__ s1,
                                   const float* __restrict__ s2,
                                   const float* __restrict__ counts,
                                   float* __restrict__ out) {
    int i = blockIdx.x * blockDim.x + threadIdx.x;
    if (i >= NB * NRINGS * NC) return;
    int c = i % NC;
    int r = (i / NC) % NRINGS;
    int b = i / (NC * NRINGS);
    float cnt  = counts[r] + 1e-6f;
    int   m    = b * NC + c;
    float mean = s1[m * NRINGS + r] / cnt;
    float msq  = s2[m * NRINGS + r] / cnt;
    float var  = msq - mean * mean;
    if (var < 0.0f) var = 0.0f;
    out[(b * NRINGS + r) * TOK + c]     = mean;
    out[(b * NRINGS + r) * TOK + 3 + c] = sqrtf(var);
}

// ---------------------------------------------------------------------------
// Pass 4: median per (b, r, c). Gather masked nonzero values into LDS once,
// then 48 rounds of value-bisection (counting rank on LDS data) to recover
// the idx-th order statistic to float precision.
// ---------------------------------------------------------------------------
__global__ __launch_bounds__(256)
void rt_median_kernel(const float* __restrict__ img,
                      const float* __restrict__ mask,
                      float* __restrict__ out) {
    __shared__ float vals[MAXRING];
    __shared__ float red[256];
    __shared__ int   scnt;
    __shared__ int   srank;

    const int tid = threadIdx.x;
    const int g   = blockIdx.x;
    const int c   = g % NC;
    const int r   = (g / NC) % NRINGS;
    const int b   = g / (NC * NRINGS);

    const float* ip = img  + (size_t)(b * NC + c) * HW;
    const float* mp = mask + (size_t)r * HW;

    if (tid == 0) scnt = 0;
    __syncthreads();

    for (int i = tid; i < HW; i += 256) {
        float mv = mp[i];
        float x  = ip[i];
        if (mv > 0.0f && x != 0.0f) {
            int pos = atomicAdd(&scnt, 1);
            if (pos < MAXRING) vals[pos] = x;
        }
    }
    __syncthreads();

    int n = scnt;
    if (n > MAXRING) n = MAXRING;
    if (n == 0) {
        if (tid == 0)
            out[(b * NRINGS + r) * TOK + 6 + c] = __builtin_inff();
        return;
    }
    int idx = (n - 1) / 2;
    if (idx < 0) idx = 0;

    // min reduction -> lo
    float lmin = 3.0e38f, lmax = -3.0e38f;
    for (int i = tid; i < n; i += 256) {
        float x = vals[i];
        lmin = fminf(lmin, x);
        lmax = fmaxf(lmax, x);
    }
    red[tid] = lmin;
    __syncthreads();
    for (int off = 128; off > 0; off >>= 1) {
        if (tid < off) red[tid] = fminf(red[tid], red[tid + off]);
        __syncthreads();
    }
    float lo = red[0];
    __syncthreads();

    // max reduction -> hi
    red[tid] = lmax;
    __syncthreads();
    for (int off = 128; off > 0; off >>= 1) {
        if (tid < off) red[tid] = fmaxf(red[tid], red[tid + off]);
        __syncthreads();
    }
    float hi = red[0];
    __syncthreads();

    // bisection on value: hi converges onto sorted[idx]
    for (int it = 0; it < 48; ++it) {
        float mid = 0.5f * (lo + hi);
        int lc = 0;
        for (int i = tid; i < n; i += 256)
            lc += (vals[i] <= mid) ? 1 : 0;
        if (tid == 0) srank = 0;
        __syncthreads();
        atomicAdd(&srank, lc);
        __syncthreads();
        int cnt = srank;
        __syncthreads();
        if (cnt >= idx + 1) hi = mid;
        else                lo = mid;
    }

    if (tid == 0)
        out[(b * NRINGS + r) * TOK + 6 + c] = hi;
}

// ---------------------------------------------------------------------------
extern "C" void kernel_launch(void* const* d_in, const int* in_sizes, int n_in,
                              void* d_out, int out_size, void* d_ws, size_t ws_size,
                              hipStream_t stream) {
    (void)in_sizes; (void)n_in; (void)out_size; (void)ws_size;

    const float* img  = (const float*)d_in[0];   // (32,3,256,256) f32
    const float* mask = (const float*)d_in[1];   // (16,256,256)   f32
    float*       out  = (float*)d_out;           // (32,16,9)      f32

    float* s1     = (float*)d_ws;                // 96*16 floats
    float* s2     = s1 + BC * NRINGS;            // 96*16 floats
    float* counts = s2 + BC * NRINGS;            // 16 floats

    // zero accumulators (graph replay safe)
    rt_zero_kernel<<<(2 * BC * NRINGS + 255) / 256, 256, 0, stream>>>(s1, 2 * BC * NRINGS);

    // WMMA moments: grid.x = K chunks, grid.y = bc tiles
    const int CHUNK = 512;
    dim3 mgrid(HW / CHUNK, BC / 16);
    rt_wmma_moments_kernel<<<mgrid, 32, 0, stream>>>(img, mask, s1, s2, CHUNK);

    // per-ring counts
    rt_counts_kernel<<<NRINGS, 256, 0, stream>>>(mask, counts);

    // mean / std
    rt_finalize_kernel<<<(NB * NRINGS * NC + 255) / 256, 256, 0, stream>>>(s1, s2, counts, out);

    // medians (one workgroup per (b,r,c))
    rt_median_kernel<<<NB * NRINGS * NC, 256, 0, stream>>>(img, mask, out);
}